// LSHSampledLayer_69965017252077
// MI455X (gfx1250) — compile-verified
//
#include <hip/hip_runtime.h>
#include <hip/hip_bf16.h>
#include <math.h>

typedef float v2f __attribute__((ext_vector_type(2)));
typedef float v8f __attribute__((ext_vector_type(8)));

#define N_ 1024
#define D_ 1024
#define S_ 32768
#define P_ 128
#define K1_ 32735u  /* int(C*(1-0.001)) : 1-indexed kth smallest */
#define K2_ 16384u  /* int(C*0.5)       : 1-indexed kth smallest */

__device__ __forceinline__ v8f wmma_f32(v2f a, v2f b, v8f c) {
  // D = A(16x4 f32) * B(4x16 f32) + C(16x16 f32)
  return __builtin_amdgcn_wmma_f32_16x16x4_f32(false, a, false, b, (short)0, c,
                                               false, false);
}

// ---------------------------------------------------------------- gather bias
__global__ void k_gather_sb(const float* __restrict__ b,
                            const int* __restrict__ sid,
                            float* __restrict__ sbg) {
  int j = blockIdx.x * 256 + threadIdx.x;
  if (j < S_) sbg[j] = b[sid[j]];
}

// ----------------------------------------- logits = x @ W[sid].T + b[sid]
// one wave per 16x64 output strip: 4 C tiles share one A fragment per K-step
// -> 1 A load + 4 B loads feed 4 v_wmma_f32_16x16x4_f32
__global__ void __launch_bounds__(256)
k_gemm_logits(const float* __restrict__ x, const float* __restrict__ W,
              const int* __restrict__ sid, const float* __restrict__ sbg,
              float* __restrict__ out) {
  const int lane = threadIdx.x & 31;
  const int wave = threadIdx.x >> 5;
  const int strip = blockIdx.x * 8 + wave;  // (N/16)*(S/64) strips
  const int m0 = (strip >> 9) << 4;         // strip / 512
  const int s0 = (strip & 511) << 6;        // 64-wide strip in S
  const int half = lane >> 4;
  const int l16 = lane & 15;
  const int koff = half << 1;

  const float* ap = x + (size_t)(m0 + l16) * D_ + koff;
  const float* bp[4];
#pragma unroll
  for (int t = 0; t < 4; ++t) {
    const int wrow = sid[s0 + t * 16 + l16];
    bp[t] = W + (size_t)wrow * D_ + koff;
  }

  v8f c0 = {}, c1 = {}, c2 = {}, c3 = {};
#pragma unroll 4
  for (int k = 0; k < D_; k += 4) {
    v2f a = *(const v2f*)(ap + k);
    v2f b0 = *(const v2f*)(bp[0] + k);
    v2f b1 = *(const v2f*)(bp[1] + k);
    v2f b2 = *(const v2f*)(bp[2] + k);
    v2f b3 = *(const v2f*)(bp[3] + k);
    c0 = wmma_f32(a, b0, c0);
    c1 = wmma_f32(a, b1, c1);
    c2 = wmma_f32(a, b2, c2);
    c3 = wmma_f32(a, b3, c3);
  }

  v8f cc[4] = {c0, c1, c2, c3};
#pragma unroll
  for (int t = 0; t < 4; ++t) {
    const int col = s0 + t * 16 + l16;
    const float bias = sbg[col];
    float* op = out + (size_t)(m0 + (half << 3)) * S_ + col;
#pragma unroll
    for (int r = 0; r < 8; ++r) op[(size_t)r * S_] = cc[t][r] + bias;
  }
}

// ------------------------- projections: out[rows,P] = [A | last] @ rp(D+1,P)
// one wave per 16x128 row strip (8 C tiles): the large gathered A stream is
// read exactly once; rp is small and L2-resident.
// sid==nullptr -> identity rows; lastcol==nullptr -> last column is 1.0
__global__ void __launch_bounds__(256)
k_proj(const float* __restrict__ A, const float* __restrict__ lastcol,
       const int* __restrict__ sid, const float* __restrict__ rp,
       float* __restrict__ outp) {
  const int lane = threadIdx.x & 31;
  const int wave = threadIdx.x >> 5;
  const int strip = blockIdx.x * 8 + wave;  // rows/16 strips
  const int r0 = strip << 4;
  const int half = lane >> 4;
  const int l16 = lane & 15;
  const int koff = half << 1;

  const int arow = r0 + l16;
  const int prow = sid ? sid[arow] : arow;
  const float* ap = A + (size_t)prow * D_ + koff;
  const float* bbase = rp + (size_t)koff * P_ + l16;

  v8f c[8] = {};
#pragma unroll 2
  for (int k = 0; k < D_; k += 4) {
    v2f a = *(const v2f*)(ap + k);
#pragma unroll
    for (int t = 0; t < 8; ++t) {
      v2f bb;
      bb.x = bbase[(size_t)k * P_ + t * 16];
      bb.y = bbase[(size_t)(k + 1) * P_ + t * 16];
      c[t] = wmma_f32(a, bb, c[t]);
    }
  }
  // tail: K = 1024 (the appended column), zero-pad K=1025..1027
  {
    const float lastv = lastcol ? lastcol[arow] : 1.0f;
    v2f a;
    a.x = (koff == 0) ? lastv : 0.0f;
    a.y = 0.0f;
#pragma unroll
    for (int t = 0; t < 8; ++t) {
      v2f bb;
      bb.x = (koff == 0) ? rp[(size_t)D_ * P_ + t * 16 + l16] : 0.0f;
      bb.y = 0.0f;
      c[t] = wmma_f32(a, bb, c[t]);
    }
  }

#pragma unroll
  for (int t = 0; t < 8; ++t) {
    float* op = outp + (size_t)(r0 + (half << 3)) * P_ + t * 16 + l16;
#pragma unroll
    for (int r = 0; r < 8; ++r) op[(size_t)r * P_] = c[t][r];
  }
}

// ------------------------------------------- row L2-normalize, rows of P=128
__global__ void __launch_bounds__(256)
k_norm(float* __restrict__ v, int rows) {
  const int row = blockIdx.x * 8 + (threadIdx.x >> 5);
  const int lane = threadIdx.x & 31;
  if (row >= rows) return;
  float4* p = (float4*)(v + (size_t)row * P_);
  float4 d = p[lane];
  float ss = d.x * d.x + d.y * d.y + d.z * d.z + d.w * d.w;
#pragma unroll
  for (int off = 16; off; off >>= 1) ss += __shfl_xor(ss, off, 32);
  const float inv = 1.0f / fmaxf(sqrtf(ss), 1e-6f);
  d.x *= inv; d.y *= inv; d.z *= inv; d.w *= inv;
  p[lane] = d;
}

// ----------------- per-row radix select (kth smallest) on product bits
__global__ void __launch_bounds__(256)
k_select(const float* __restrict__ logits, const float* __restrict__ sbg,
         float* __restrict__ kth1, float* __restrict__ kth2) {
  __shared__ unsigned hist[256];
  __shared__ unsigned sh_sel, sh_k;
  const int row = blockIdx.x;
  const int tid = threadIdx.x;
  const float* lrow = logits + (size_t)row * S_;
  const unsigned kt[2] = {K1_, K2_};

  for (int t = 0; t < 2; ++t) {
    unsigned prefix = 0, pmask = 0, k = kt[t];
#pragma unroll 1
    for (int pass = 0; pass < 4; ++pass) {
      const int shift = 24 - 8 * pass;
      hist[tid] = 0;
      __syncthreads();
      for (int j = tid; j < S_; j += 256) {
        const float vv = lrow[j] - sbg[j];
        unsigned u = __float_as_uint(vv);
        u ^= (u & 0x80000000u) ? 0xFFFFFFFFu : 0x80000000u;  // order-preserving
        if ((u & pmask) == prefix) atomicAdd(&hist[(u >> shift) & 255u], 1u);
      }
      __syncthreads();
      if (tid == 0) {
        unsigned cum = 0, sel = 255, kk = k;
        for (int bkt = 0; bkt < 256; ++bkt) {
          const unsigned cc = hist[bkt];
          if (cum + cc >= kk) { sel = (unsigned)bkt; kk -= cum; break; }
          cum += cc;
        }
        sh_sel = sel;
        sh_k = kk;
      }
      __syncthreads();
      prefix |= sh_sel << shift;
      pmask |= 255u << shift;
      k = sh_k;
      __syncthreads();
    }
    if (tid == 0) {
      unsigned u = prefix;
      u = (u & 0x80000000u) ? (u ^ 0x80000000u) : ~u;  // inverse map
      (t == 0 ? kth1 : kth2)[row] = __uint_as_float(u);
    }
    __syncthreads();
  }
}

// -------------------------- thresholds = mean over rows; zero accumulators
__global__ void __launch_bounds__(256)
k_thresh(const float* __restrict__ kth1, const float* __restrict__ kth2,
         float* __restrict__ scal, float* __restrict__ accum) {
  __shared__ float s1[256], s2[256];
  const int tid = threadIdx.x;
  float a = 0.0f, b = 0.0f;
  for (int j = tid; j < N_; j += 256) { a += kth1[j]; b += kth2[j]; }
  s1[tid] = a; s2[tid] = b;
  __syncthreads();
  for (int off = 128; off; off >>= 1) {
    if (tid < off) { s1[tid] += s1[tid + off]; s2[tid] += s2[tid + off]; }
    __syncthreads();
  }
  if (tid == 0) {
    scal[0] = fmaxf(s1[0] / (float)N_, 0.0f);  // t1_ip
    scal[1] = s2[0] / (float)N_;               // t2_ip
    accum[0] = 0.0f; accum[1] = 0.0f; accum[2] = 0.0f; accum[3] = 0.0f;
  }
}

// ---------------- sim = qn @ wn.T (K=128) fused with masked accumulation
// one wave per 16x64 strip (4 C tiles sharing the A fragment)
__global__ void __launch_bounds__(256)
k_sim(const float* __restrict__ qn, const float* __restrict__ wn,
      const float* __restrict__ logits, const float* __restrict__ sbg,
      const float* __restrict__ scal, float* __restrict__ accum) {
  const int lane = threadIdx.x & 31;
  const int wave = threadIdx.x >> 5;
  const int strip = blockIdx.x * 8 + wave;
  const int m0 = (strip >> 9) << 4;
  const int s0 = (strip & 511) << 6;
  const int half = lane >> 4;
  const int l16 = lane & 15;
  const int koff = half << 1;

  const float* ap = qn + (size_t)(m0 + l16) * P_ + koff;
  const float* bp[4];
#pragma unroll
  for (int t = 0; t < 4; ++t)
    bp[t] = wn + (size_t)(s0 + t * 16 + l16) * P_ + koff;

  v8f c0 = {}, c1 = {}, c2 = {}, c3 = {};
#pragma unroll 4
  for (int k = 0; k < P_; k += 4) {
    v2f a = *(const v2f*)(ap + k);
    v2f b0 = *(const v2f*)(bp[0] + k);
    v2f b1 = *(const v2f*)(bp[1] + k);
    v2f b2 = *(const v2f*)(bp[2] + k);
    v2f b3 = *(const v2f*)(bp[3] + k);
    c0 = wmma_f32(a, b0, c0);
    c1 = wmma_f32(a, b1, c1);
    c2 = wmma_f32(a, b2, c2);
    c3 = wmma_f32(a, b3, c3);
  }

  const float t1 = scal[0];
  const float t2 = scal[1];
  const int mbase = m0 + (half << 3);

  v8f cc[4] = {c0, c1, c2, c3};
  float sp = 0.0f, cp = 0.0f, sn = 0.0f, cn = 0.0f;
#pragma unroll
  for (int t = 0; t < 4; ++t) {
    const int j = s0 + t * 16 + l16;
    const float sb_j = sbg[j];
    const float* lp = logits + (size_t)mbase * S_ + j;
#pragma unroll
    for (int r = 0; r < 8; ++r) {
      const float prod = lp[(size_t)r * S_] - sb_j;
      const float s = cc[t][r];
      if (prod > t1) { sp += s; cp += 1.0f; }
      if (prod < t2) { sn += s; cn += 1.0f; }
    }
  }
#pragma unroll
  for (int off = 16; off; off >>= 1) {
    sp += __shfl_down(sp, off, 32);
    cp += __shfl_down(cp, off, 32);
    sn += __shfl_down(sn, off, 32);
    cn += __shfl_down(cn, off, 32);
  }
  __shared__ float red[8][4];
  if (lane == 0) {
    red[wave][0] = sp; red[wave][1] = cp; red[wave][2] = sn; red[wave][3] = cn;
  }
  __syncthreads();
  if (threadIdx.x == 0) {
    float a0 = 0, a1 = 0, a2 = 0, a3 = 0;
#pragma unroll
    for (int w = 0; w < 8; ++w) {
      a0 += red[w][0]; a1 += red[w][1]; a2 += red[w][2]; a3 += red[w][3];
    }
    atomicAdd(accum + 0, a0);
    atomicAdd(accum + 1, a1);
    atomicAdd(accum + 2, a2);
    atomicAdd(accum + 3, a3);
  }
}

// ------------------------------------------------------------- final hinge
__global__ void k_final(const float* __restrict__ accum,
                        float* __restrict__ out_loss) {
  const float simp = accum[0] / fmaxf(accum[1], 1.0f);
  const float simn = accum[2] / fmaxf(accum[3], 1.0f);
  out_loss[0] = fmaxf(simn - simp + 1.0f, 0.0f);
}

extern "C" void kernel_launch(void* const* d_in, const int* in_sizes, int n_in,
                              void* d_out, int out_size, void* d_ws,
                              size_t ws_size, hipStream_t stream) {
  (void)in_sizes; (void)n_in; (void)out_size; (void)ws_size;
  const float* x  = (const float*)d_in[0];
  const float* W  = (const float*)d_in[1];
  const float* b  = (const float*)d_in[2];
  const float* rp = (const float*)d_in[3];
  const int* sid  = (const int*)d_in[4];

  float* out = (float*)d_out;  // [N*S logits][1 loss]
  float* ws  = (float*)d_ws;
  float* sbg  = ws;                     // S
  float* kth1 = sbg + S_;               // N
  float* kth2 = kth1 + N_;              // N
  float* scal = kth2 + N_;              // 2 (t1_ip, t2_ip)
  float* accum = scal + 2;              // 4
  float* qn = scal + 8;                 // N*P
  float* wn = qn + (size_t)N_ * P_;     // S*P

  k_gather_sb<<<S_ / 256, 256, 0, stream>>>(b, sid, sbg);
  // (N/16)*(S/64) strips, 8 waves per block
  k_gemm_logits<<<(N_ / 16) * (S_ / 64) / 8, 256, 0, stream>>>(x, W, sid, sbg,
                                                               out);
  k_proj<<<(S_ / 16) / 8, 256, 0, stream>>>(W, sbg, sid, rp, wn);
  k_proj<<<(N_ / 16) / 8, 256, 0, stream>>>(x, nullptr, nullptr, rp, qn);
  k_norm<<<S_ / 8, 256, 0, stream>>>(wn, S_);
  k_norm<<<N_ / 8, 256, 0, stream>>>(qn, N_);
  k_select<<<N_, 256, 0, stream>>>(out, sbg, kth1, kth2);
  k_thresh<<<1, 256, 0, stream>>>(kth1, kth2, scal, accum);
  k_sim<<<(N_ / 16) * (S_ / 64) / 8, 256, 0, stream>>>(qn, wn, out, sbg, scal,
                                                       accum);
  k_final<<<1, 1, 0, stream>>>(accum, out + (size_t)N_ * S_);
}